// GCNModel_27719718928688
// MI455X (gfx1250) — compile-verified
//
#include <hip/hip_runtime.h>
#include <hip/hip_bf16.h>

#define GCN_N 100000
#define GCN_E 800000
#define GCN_H 128
#define GCN_C 2
#define BN_EPS 1e-5f

#define MM_ROWS 64                                   // rows per block (4 M-tiles per wave)
#define MM_TILES (MM_ROWS / 16)
#define MM_BLOCKS ((GCN_N + MM_ROWS - 1) / MM_ROWS)  // 1563
#define GCN_NPAD (MM_BLOCKS * MM_ROWS)               // 100032: padded row count

typedef __attribute__((ext_vector_type(2))) float v2f;
typedef __attribute__((ext_vector_type(8))) float v8f;

// ---------------------------------------------------------------------------
// Degree: atomic count of incoming edges per node (deg buffer pre-zeroed)
// ---------------------------------------------------------------------------
__global__ __launch_bounds__(256) void deg_kernel(const long long* __restrict__ ei,
                                                  float* __restrict__ deg) {
    int i = blockIdx.x * 256 + threadIdx.x;
    if (i >= GCN_E) return;
    int d = (int)ei[(size_t)GCN_E + i];   // dst row
    atomicAdd(&deg[d], 1.0f);
}

// dis = rsqrt(deg + 1) (in place over deg), self_norm = dis*dis
__global__ __launch_bounds__(256) void dis_kernel(float* __restrict__ deg_dis,
                                                  float* __restrict__ self_norm) {
    int i = blockIdx.x * 256 + threadIdx.x;
    if (i >= GCN_N) return;
    float dis = rsqrtf(deg_dis[i] + 1.0f);
    deg_dis[i] = dis;
    self_norm[i] = dis * dis;
}

// norm[e] = dis[src] * dis[dst]
__global__ __launch_bounds__(256) void norm_kernel(const long long* __restrict__ ei,
                                                   const float* __restrict__ dis,
                                                   float* __restrict__ norm) {
    int e = blockIdx.x * 256 + threadIdx.x;
    if (e >= GCN_E) return;
    int s = (int)ei[e];
    int d = (int)ei[(size_t)GCN_E + e];
    norm[e] = dis[s] * dis[d];
}

// ---------------------------------------------------------------------------
// hw = A(N x 128) @ W(128 x 128) using v_wmma_f32_16x16x4_f32.
// Block = 256 threads = 8 waves; block computes a 64-row x 128-col tile.
// Wave w owns output columns [16w, 16w+16) and 4 M-tiles, so each B fragment
// (one global load pair per K-step) is reused by 4 WMMAs.
// The output buffer has GCN_NPAD rows, so D stores are always unguarded; only
// the A staging read is guarded (A may be the exactly-sized input x).
// ---------------------------------------------------------------------------
__global__ __launch_bounds__(256) void gemm128_wmma(const float* __restrict__ A,
                                                    const float* __restrict__ W,
                                                    float* __restrict__ out) {
    __shared__ float As[MM_ROWS][GCN_H];    // 32 KB
    const int bm  = blockIdx.x * MM_ROWS;
    const int tid = threadIdx.x;

    // Stage 64x128 f32 A-tile into LDS; zero-fill rows past N (last block only).
    if ((bm + MM_ROWS) <= GCN_N) {
        for (int i = tid; i < MM_ROWS * GCN_H / 4; i += 256) {
            int r = (i * 4) / GCN_H;
            int c = (i * 4) % GCN_H;
            *(float4*)&As[r][c] = *(const float4*)&A[(size_t)(bm + r) * GCN_H + c];
        }
    } else {
        for (int i = tid; i < MM_ROWS * GCN_H / 4; i += 256) {
            int r = (i * 4) / GCN_H;
            int c = (i * 4) % GCN_H;
            float4 v = make_float4(0.f, 0.f, 0.f, 0.f);
            if (bm + r < GCN_N)
                v = *(const float4*)&A[(size_t)(bm + r) * GCN_H + c];
            *(float4*)&As[r][c] = v;
        }
    }
    __syncthreads();

    const int wave = tid >> 5;        // 0..7 -> 16-wide column tile
    const int lane = tid & 31;
    const int half = lane >> 4;       // 0: lanes 0-15, 1: lanes 16-31
    const int l16  = lane & 15;
    const int col  = wave * 16 + l16;

    v8f acc[MM_TILES];
    #pragma unroll
    for (int t = 0; t < MM_TILES; ++t) acc[t] = (v8f){};

    for (int kk = 0; kk < GCN_H; kk += 4) {
        const int k0 = kk + 2 * half;                 // K pair owned by this half
        v2f b;
        b.x = W[(size_t)k0 * GCN_H + col];            // B col = col
        b.y = W[(size_t)(k0 + 1) * GCN_H + col];
        #pragma unroll
        for (int t = 0; t < MM_TILES; ++t) {
            v2f a;
            a.x = As[t * 16 + l16][k0];               // A row = t*16 + l16
            a.y = As[t * 16 + l16][k0 + 1];
            acc[t] = __builtin_amdgcn_wmma_f32_16x16x4_f32(
                /*neg_a=*/false, a, /*neg_b=*/false, b,
                /*c_mod=*/(short)0, acc[t], /*reuse_a=*/false, /*reuse_b=*/false);
        }
    }

    // D layout: VGPR r -> (M = t*16 + r + 8*half, N = col). out has NPAD rows,
    // so no guards: padded rows receive (zero) garbage that is never consumed.
    #pragma unroll
    for (int t = 0; t < MM_TILES; ++t) {
        #pragma unroll
        for (int r = 0; r < 8; ++r) {
            int row = bm + t * 16 + r + 8 * half;
            out[(size_t)row * GCN_H + col] = acc[t][r];
        }
    }
}

// ---------------------------------------------------------------------------
// Edge scatter: agg[dst] += hw[src] * norm[e].  One thread per (edge, 4 feats).
// hw is L2-resident (51 MB < 192 MB L2) so the random gather hits L2.
// ---------------------------------------------------------------------------
__global__ __launch_bounds__(256) void scatter_kernel(const long long* __restrict__ ei,
                                                      const float* __restrict__ norm,
                                                      const float* __restrict__ hw,
                                                      float* __restrict__ agg) {
    long long idx = (long long)blockIdx.x * 256 + threadIdx.x;
    if (idx >= (long long)GCN_E * 32) return;
    int e = (int)(idx >> 5);
    int f = (int)(idx & 31) * 4;
    int s = (int)ei[e];
    int d = (int)ei[(size_t)GCN_E + e];
    float nv = norm[e];
    float4 v = *(const float4*)&hw[(size_t)s * GCN_H + f];
    float* ap = &agg[(size_t)d * GCN_H + f];
    atomicAdd(ap + 0, v.x * nv);
    atomicAdd(ap + 1, v.y * nv);
    atomicAdd(ap + 2, v.z * nv);
    atomicAdd(ap + 3, v.w * nv);
}

// ---------------------------------------------------------------------------
// Epilogues (float4-vectorized: pure bandwidth passes over ~51 MB buffers)
// ---------------------------------------------------------------------------
// layer 0: out = relu(agg + hw*self_norm + bias)
__global__ __launch_bounds__(256) void combine_relu(const float* __restrict__ agg,
                                                    const float* __restrict__ hw,
                                                    const float* __restrict__ self_norm,
                                                    const float* __restrict__ bias,
                                                    float* __restrict__ out) {
    int q = blockIdx.x * 256 + threadIdx.x;           // quad index over N*H/4
    if (q >= GCN_N * GCN_H / 4) return;
    int idx = q * 4;
    int i = idx >> 7;
    int f = idx & 127;
    float  sn = self_norm[i];
    float4 a  = *(const float4*)&agg[idx];
    float4 hv = *(const float4*)&hw[idx];
    float4 bv = *(const float4*)&bias[f];
    float4 o;
    o.x = fmaxf(a.x + hv.x * sn + bv.x, 0.0f);
    o.y = fmaxf(a.y + hv.y * sn + bv.y, 0.0f);
    o.z = fmaxf(a.z + hv.z * sn + bv.z, 0.0f);
    o.w = fmaxf(a.w + hv.w * sn + bv.w, 0.0f);
    *(float4*)&out[idx] = o;
}

// layers 1/2: out = relu(bn(agg + hw*self_norm + bias)) + res
__global__ __launch_bounds__(256) void combine_bn_relu_res(const float* __restrict__ agg,
                                                           const float* __restrict__ hw,
                                                           const float* __restrict__ self_norm,
                                                           const float* __restrict__ bias,
                                                           const float* __restrict__ bn_g,
                                                           const float* __restrict__ bn_b,
                                                           const float* __restrict__ bn_m,
                                                           const float* __restrict__ bn_v,
                                                           const float* __restrict__ res,
                                                           float* __restrict__ out) {
    int q = blockIdx.x * 256 + threadIdx.x;
    if (q >= GCN_N * GCN_H / 4) return;
    int idx = q * 4;
    int i = idx >> 7;
    int f = idx & 127;
    float  sn = self_norm[i];
    float4 a  = *(const float4*)&agg[idx];
    float4 hv = *(const float4*)&hw[idx];
    float4 bv = *(const float4*)&bias[f];
    float4 g  = *(const float4*)&bn_g[f];
    float4 bb = *(const float4*)&bn_b[f];
    float4 m  = *(const float4*)&bn_m[f];
    float4 vv = *(const float4*)&bn_v[f];
    float4 rv = *(const float4*)&res[idx];
    float4 o;
    float t;
    t   = a.x + hv.x * sn + bv.x;
    o.x = fmaxf((t - m.x) * (g.x * rsqrtf(vv.x + BN_EPS)) + bb.x, 0.0f) + rv.x;
    t   = a.y + hv.y * sn + bv.y;
    o.y = fmaxf((t - m.y) * (g.y * rsqrtf(vv.y + BN_EPS)) + bb.y, 0.0f) + rv.y;
    t   = a.z + hv.z * sn + bv.z;
    o.z = fmaxf((t - m.z) * (g.z * rsqrtf(vv.z + BN_EPS)) + bb.z, 0.0f) + rv.z;
    t   = a.w + hv.w * sn + bv.w;
    o.w = fmaxf((t - m.w) * (g.w * rsqrtf(vv.w + BN_EPS)) + bb.w, 0.0f) + rv.w;
    *(float4*)&out[idx] = o;
}

// Output head: hw3[i][:] = h[i] @ W3 (128x2), one thread per node, float4 loads.
// A float4 of row-major W3 covers 2 K values for both classes.
__global__ __launch_bounds__(256) void gemm_out(const float* __restrict__ h,
                                                const float* __restrict__ W3,
                                                float* __restrict__ hw3) {
    int i = blockIdx.x * 256 + threadIdx.x;
    if (i >= GCN_N) return;
    const float* hp = &h[(size_t)i * GCN_H];
    float acc0 = 0.0f, acc1 = 0.0f;
    #pragma unroll 8
    for (int k = 0; k < GCN_H; k += 4) {
        float4 hv  = *(const float4*)&hp[k];
        float4 w01 = *(const float4*)&W3[k * GCN_C];         // k, k+1
        float4 w23 = *(const float4*)&W3[(k + 2) * GCN_C];   // k+2, k+3
        acc0 += hv.x * w01.x + hv.y * w01.z + hv.z * w23.x + hv.w * w23.z;
        acc1 += hv.x * w01.y + hv.y * w01.w + hv.z * w23.y + hv.w * w23.w;
    }
    hw3[(size_t)i * GCN_C + 0] = acc0;
    hw3[(size_t)i * GCN_C + 1] = acc1;
}

// Edge scatter for C=2: one thread per edge
__global__ __launch_bounds__(256) void scatter_out(const long long* __restrict__ ei,
                                                   const float* __restrict__ norm,
                                                   const float* __restrict__ hw3,
                                                   float* __restrict__ agg3) {
    int e = blockIdx.x * 256 + threadIdx.x;
    if (e >= GCN_E) return;
    int s = (int)ei[e];
    int d = (int)ei[(size_t)GCN_E + e];
    float nv = norm[e];
    atomicAdd(&agg3[(size_t)d * GCN_C + 0], hw3[(size_t)s * GCN_C + 0] * nv);
    atomicAdd(&agg3[(size_t)d * GCN_C + 1], hw3[(size_t)s * GCN_C + 1] * nv);
}

// z = agg3 + hw3*self_norm + b3; out = log_softmax(z) over 2 classes
__global__ __launch_bounds__(256) void final_kernel(const float* __restrict__ agg3,
                                                    const float* __restrict__ hw3,
                                                    const float* __restrict__ self_norm,
                                                    const float* __restrict__ b3,
                                                    float* __restrict__ out) {
    int i = blockIdx.x * 256 + threadIdx.x;
    if (i >= GCN_N) return;
    float sn = self_norm[i];
    float z0 = agg3[(size_t)i * 2 + 0] + hw3[(size_t)i * 2 + 0] * sn + b3[0];
    float z1 = agg3[(size_t)i * 2 + 1] + hw3[(size_t)i * 2 + 1] * sn + b3[1];
    float mx  = fmaxf(z0, z1);
    float lse = mx + logf(expf(z0 - mx) + expf(z1 - mx));
    out[(size_t)i * 2 + 0] = z0 - lse;
    out[(size_t)i * 2 + 1] = z1 - lse;
}

// ---------------------------------------------------------------------------
extern "C" void kernel_launch(void* const* d_in, const int* in_sizes, int n_in,
                              void* d_out, int out_size, void* d_ws, size_t ws_size,
                              hipStream_t stream) {
    (void)in_sizes; (void)n_in; (void)out_size; (void)ws_size;

    const float*     x    = (const float*)d_in[0];
    const long long* ei   = (const long long*)d_in[1];
    const float*     W0   = (const float*)d_in[2];
    const float*     b0   = (const float*)d_in[3];
    const float*     W1   = (const float*)d_in[4];
    const float*     b1   = (const float*)d_in[5];
    const float*     W2   = (const float*)d_in[6];
    const float*     b2   = (const float*)d_in[7];
    const float*     W3   = (const float*)d_in[8];
    const float*     b3   = (const float*)d_in[9];
    const float*     bn0g = (const float*)d_in[10];
    const float*     bn0b = (const float*)d_in[11];
    const float*     bn0m = (const float*)d_in[12];
    const float*     bn0v = (const float*)d_in[13];
    const float*     bn1g = (const float*)d_in[14];
    const float*     bn1b = (const float*)d_in[15];
    const float*     bn1m = (const float*)d_in[16];
    const float*     bn1v = (const float*)d_in[17];
    float* out = (float*)d_out;

    // workspace layout (floats); hw/hA/hB padded to GCN_NPAD rows so the WMMA
    // GEMM can store full 64-row tiles without guards.
    float* w = (float*)d_ws;
    size_t off = 0;
    float* dis       = w + off; off += GCN_N;                  // deg -> dis in place
    float* self_norm = w + off; off += GCN_N;
    float* norm      = w + off; off += GCN_E;
    float* hw        = w + off; off += (size_t)GCN_NPAD * GCN_H;
    float* agg       = w + off; off += (size_t)GCN_N * GCN_H;
    float* hA        = w + off; off += (size_t)GCN_NPAD * GCN_H;
    float* hB        = w + off; off += (size_t)GCN_NPAD * GCN_H;
    float* hw3  = hw;    // reuse (head runs after last 128-wide use of hw)
    float* agg3 = agg;   // reuse

    const int gridE   = (GCN_E + 255) / 256;                     // per-edge
    const int gridN   = (GCN_N + 255) / 256;                     // per-node
    const int gridNH4 = (GCN_N * GCN_H / 4 + 255) / 256;         // per-float4
    const int gridSc  = (int)(((long long)GCN_E * 32 + 255) / 256);

    // normalization
    hipMemsetAsync(dis, 0, (size_t)GCN_N * sizeof(float), stream);
    deg_kernel <<<gridE, 256, 0, stream>>>(ei, dis);
    dis_kernel <<<gridN, 256, 0, stream>>>(dis, self_norm);
    norm_kernel<<<gridE, 256, 0, stream>>>(ei, dis, norm);

    // layer 0: hA = relu(conv(x, W0, b0))
    gemm128_wmma<<<MM_BLOCKS, 256, 0, stream>>>(x, W0, hw);
    hipMemsetAsync(agg, 0, (size_t)GCN_N * GCN_H * sizeof(float), stream);
    scatter_kernel<<<gridSc, 256, 0, stream>>>(ei, norm, hw, agg);
    combine_relu<<<gridNH4, 256, 0, stream>>>(agg, hw, self_norm, b0, hA);

    // layer 1: hB = relu(bn0(conv(hA, W1, b1))) + hA
    gemm128_wmma<<<MM_BLOCKS, 256, 0, stream>>>(hA, W1, hw);
    hipMemsetAsync(agg, 0, (size_t)GCN_N * GCN_H * sizeof(float), stream);
    scatter_kernel<<<gridSc, 256, 0, stream>>>(ei, norm, hw, agg);
    combine_bn_relu_res<<<gridNH4, 256, 0, stream>>>(agg, hw, self_norm, b1,
                                                     bn0g, bn0b, bn0m, bn0v, hA, hB);

    // layer 2: hA = relu(bn1(conv(hB, W2, b2))) + hB
    gemm128_wmma<<<MM_BLOCKS, 256, 0, stream>>>(hB, W2, hw);
    hipMemsetAsync(agg, 0, (size_t)GCN_N * GCN_H * sizeof(float), stream);
    scatter_kernel<<<gridSc, 256, 0, stream>>>(ei, norm, hw, agg);
    combine_bn_relu_res<<<gridNH4, 256, 0, stream>>>(agg, hw, self_norm, b2,
                                                     bn1g, bn1b, bn1m, bn1v, hB, hA);

    // output head: out = log_softmax(conv(hA, W3, b3))
    gemm_out<<<gridN, 256, 0, stream>>>(hA, W3, hw3);
    hipMemsetAsync(agg3, 0, (size_t)GCN_N * GCN_C * sizeof(float), stream);
    scatter_out<<<gridE, 256, 0, stream>>>(ei, norm, hw3, agg3);
    final_kernel<<<gridN, 256, 0, stream>>>(agg3, hw3, self_norm, b3, out);
}